// GNS_12360915878662
// MI455X (gfx1250) — compile-verified
//
#include <hip/hip_runtime.h>
#include <hip/hip_bf16.h>
#include <math.h>

// ---------------------------------------------------------------------------
// GNS-style GNN forward for MI455X (gfx1250), fp32 end-to-end using
// V_WMMA_F32_16X16X4_F32.  One wave (32 threads) per block; the edge kernel
// processes 2 row-tiles (32 edges) per wave so each B-fragment load feeds
// 2 WMMAs.  Weights are pre-packed (zero-padded, k-pair interleaved) so the
// inner loop has zero conditionals: 1 coalesced global_load_b64 per B-frag.
// ELU uses the v_exp_f32 hardware transcendental (branchless epilogue).
// ---------------------------------------------------------------------------

typedef float v2f __attribute__((ext_vector_type(2)));
typedef float v8f __attribute__((ext_vector_type(8)));

#if defined(__HIP_DEVICE_COMPILE__)
#if !__has_builtin(__builtin_amdgcn_wmma_f32_16x16x4_f32)
#error "missing __builtin_amdgcn_wmma_f32_16x16x4_f32 on this toolchain"
#endif
#endif

#define N_NODES 50000
#define N_EDGES 800000
#define EDGE_R  2          // row-tiles (of 16 edges) per wave in k_edge

__device__ __forceinline__ float elu1(float x) {
  // fast hw exp: v_mul + v_exp_f32; ternary if-converts to v_cndmask
  return x > 0.0f ? x : (__expf(x) - 1.0f);
}

// ---------------------------------------------------------------------------
// C[rt] = A_rt(16 x FAN_IN, LDS, leading dim LDA, zero-padded to 4*K4)
//         @ W(FAN_IN x 64) given as packed Wp:
//   Wp[(kt*2 + half)*64 + col] = { W[2*(kt*2+half)][col], W[2*(kt*2+half)+1][col] }
//   (zero-padded past FAN_IN), so lane (half=lane>>4) loads its K-pair as one b64.
// A-frag (ISA 7.12.2, 32-bit A 16x4): lanes 0-15 K={k0,k0+1}, lanes 16-31 K={k0+2,k0+3}.
// D: VGPR r -> row r (lanes 0-15) / r+8 (lanes 16-31), col = (lane&15) in 16-col tile.
// ---------------------------------------------------------------------------
template <int FAN_IN, int LDA, int R>
__device__ __forceinline__ void mm16x64_mt(const float* At, int tstride,
                                           const v2f* __restrict__ Wp,
                                           v8f (&acc)[R][4]) {
  const int lane = threadIdx.x & 31;
  const int m    = lane & 15;
  const int half = lane >> 4;
  v8f z = {};
#pragma unroll
  for (int rt = 0; rt < R; ++rt)
#pragma unroll
    for (int nt = 0; nt < 4; ++nt) acc[rt][nt] = z;
  constexpr int K4 = (FAN_IN + 3) / 4;
  for (int kt = 0; kt < K4; ++kt) {
    const int k0 = kt * 4 + half * 2;
    v2f b[4];
#pragma unroll
    for (int nt = 0; nt < 4; ++nt)
      b[nt] = Wp[(kt * 2 + half) * 64 + nt * 16 + m];
#pragma unroll
    for (int rt = 0; rt < R; ++rt) {
      const float* Ab = At + rt * tstride;
      v2f a;
      a.x = Ab[m * LDA + k0];
      a.y = Ab[m * LDA + k0 + 1];
#pragma unroll
      for (int nt = 0; nt < 4; ++nt)
        acc[rt][nt] = __builtin_amdgcn_wmma_f32_16x16x4_f32(
            false, a, false, b[nt], (short)0, acc[rt][nt], false, false);
    }
  }
}

__device__ __forceinline__ void epi_to_lds(const v8f (&acc)[4],
                                           const float* __restrict__ bias,
                                           float* Ht, int ldh, bool do_elu) {
  const int lane = threadIdx.x & 31;
  const int n    = lane & 15;
  const int roff = (lane >> 4) << 3;  // 0 or 8
#pragma unroll
  for (int nt = 0; nt < 4; ++nt) {
    const int col = nt * 16 + n;
    const float bv = bias[col];
#pragma unroll
    for (int r = 0; r < 8; ++r) {
      float v = acc[nt][r] + bv;
      if (do_elu) v = elu1(v);
      Ht[(r + roff) * ldh + col] = v;
    }
  }
}

// -------------------------- weight pre-packing -----------------------------
// Wp has k4*2*64 v2f entries; entry idx = pair*64 + col, pair = kt*2 + half,
// holds rows {2*pair, 2*pair+1} of W (zero if >= fan_in).
__global__ void k_pack(const float* __restrict__ W, float* __restrict__ Wp,
                       int fan_in, int k4) {
  int idx = blockIdx.x * blockDim.x + threadIdx.x;
  int tot = k4 * 2 * 64;
  if (idx >= tot) return;
  int col  = idx & 63;
  int pair = idx >> 6;
  int r0 = 2 * pair, r1 = 2 * pair + 1;
  Wp[idx * 2]     = (r0 < fan_in) ? W[r0 * 64 + col] : 0.0f;
  Wp[idx * 2 + 1] = (r1 < fan_in) ? W[r1 * 64 + col] : 0.0f;
}

// ----------------------------- edge degree ---------------------------------
__global__ void k_count(const int* __restrict__ dst, float* __restrict__ cnt) {
  int e = blockIdx.x * blockDim.x + threadIdx.x;
  if (e < N_EDGES) atomicAdd(&cnt[dst[e]], 1.0f);
}
__global__ void k_inv(float* __restrict__ cnt) {
  int n = blockIdx.x * blockDim.x + threadIdx.x;
  if (n < N_NODES) {
    float c = cnt[n];
    cnt[n] = (c > 0.0f) ? (1.0f / c) : 0.0f;
  }
}

// ------------------------------ encoder ------------------------------------
__global__ void __launch_bounds__(32)
k_encoder(const float* __restrict__ x,
          const v2f* __restrict__ pW1, const float* __restrict__ b1,
          const v2f* __restrict__ pW2, const float* __restrict__ b2,
          float* __restrict__ h) {
  __shared__ float At[16 * 20];
  __shared__ float Ht[16 * 68];
  const int lane  = threadIdx.x;
  const int node0 = blockIdx.x * 16;
  for (int i = lane; i < 16 * 20; i += 32) {
    int r = i / 20, c = i - r * 20;
    At[i] = (c < 9) ? x[(node0 + r) * 9 + c] : 0.0f;
  }
  __syncthreads();
  v8f acc[1][4];
  mm16x64_mt<9, 20, 1>(At, 0, pW1, acc);
  epi_to_lds(acc[0], b1, Ht, 68, true);
  __syncthreads();
  mm16x64_mt<64, 68, 1>(Ht, 0, pW2, acc);
  const int n = lane & 15, roff = (lane >> 4) << 3;
#pragma unroll
  for (int nt = 0; nt < 4; ++nt) {
    const int col = nt * 16 + n;
    const float bv = b2[col];
#pragma unroll
    for (int r = 0; r < 8; ++r)
      h[(node0 + r + roff) * 64 + col] = elu1(acc[0][nt][r] + bv);
  }
}

// ----------------------------- edge message --------------------------------
// m = elu([h_dst|h_src|ea] @ W1 + b1) @ W2 + b2 ; atomic scatter into aggr.
// EDGE_R row-tiles (32 edges) per wave: B-frags reused across tiles.
__global__ void __launch_bounds__(32)
k_edge(const float* __restrict__ h, const int* __restrict__ ei,
       const float* __restrict__ ea,
       const v2f* __restrict__ pW1, const float* __restrict__ b1,
       const v2f* __restrict__ pW2, const float* __restrict__ b2,
       float* __restrict__ aggr) {
  __shared__ float At[EDGE_R * 16 * 136];
  __shared__ float Ht[EDGE_R * 16 * 68];
  __shared__ int sDst[EDGE_R * 16], sSrc[EDGE_R * 16];
  const int lane = threadIdx.x;
  const int e0   = blockIdx.x * (EDGE_R * 16);
  const int* srcIdx = ei;             // edge_index[0]
  const int* dstIdx = ei + N_EDGES;   // edge_index[1]
  sDst[lane] = dstIdx[e0 + lane];     // 32 lanes cover 32 edges exactly
  sSrc[lane] = srcIdx[e0 + lane];
  __syncthreads();
  for (int i = lane; i < EDGE_R * 16 * 136; i += 32) {
    int r = i / 136, c = i - r * 136;
    float v = 0.0f;
    if (c < 64)       v = h[sDst[r] * 64 + c];            // x_i (target)
    else if (c < 128) v = h[sSrc[r] * 64 + (c - 64)];     // x_j (source)
    else if (c < 131) v = ea[(e0 + r) * 3 + (c - 128)];
    At[i] = v;
  }
  __syncthreads();
  v8f acc[EDGE_R][4];
  mm16x64_mt<131, 136, EDGE_R>(At, 16 * 136, pW1, acc);
#pragma unroll
  for (int rt = 0; rt < EDGE_R; ++rt)
    epi_to_lds(acc[rt], b1, Ht + rt * 16 * 68, 68, true);
  __syncthreads();
  mm16x64_mt<64, 68, EDGE_R>(Ht, 16 * 68, pW2, acc);
  const int n = lane & 15, roff = (lane >> 4) << 3;
#pragma unroll
  for (int rt = 0; rt < EDGE_R; ++rt)
#pragma unroll
    for (int nt = 0; nt < 4; ++nt) {
      const int col = nt * 16 + n;
      const float bv = b2[col];
#pragma unroll
      for (int r = 0; r < 8; ++r)
        atomicAdd(&aggr[sDst[rt * 16 + r + roff] * 64 + col],
                  acc[rt][nt][r] + bv);
    }
}

// ------------------------- node update + GroupNorm -------------------------
__global__ void __launch_bounds__(32)
k_update(const float* __restrict__ h, const float* __restrict__ aggr,
         const float* __restrict__ inv,
         const v2f* __restrict__ pW1, const float* __restrict__ b1,
         const v2f* __restrict__ pW2, const float* __restrict__ b2,
         const float* __restrict__ gnw, const float* __restrict__ gnb,
         float* __restrict__ hout) {
  __shared__ float At[16 * 132];
  __shared__ float Ht[16 * 68];
  __shared__ float Gt[16 * 68];
  const int lane  = threadIdx.x;
  const int node0 = blockIdx.x * 16;
  for (int i = lane; i < 16 * 132; i += 32) {
    int r = i / 132, c = i - r * 132;
    int node = node0 + r;
    float v = 0.0f;
    if (c < 64)       v = h[node * 64 + c];
    else if (c < 128) v = aggr[node * 64 + (c - 64)] * inv[node];
    At[i] = v;
  }
  __syncthreads();
  v8f acc[1][4];
  mm16x64_mt<128, 132, 1>(At, 0, pW1, acc);
  epi_to_lds(acc[0], b1, Ht, 68, true);
  __syncthreads();
  mm16x64_mt<64, 68, 1>(Ht, 0, pW2, acc);
  epi_to_lds(acc[0], b2, Gt, 68, true);   // bias + ELU into Gt
  __syncthreads();
  // GroupNorm: 2 groups of 32 channels; lane -> (row = lane&15, grp = lane>>4)
  {
    const int row = lane & 15, grp = lane >> 4;
    const float* g = &Gt[row * 68 + grp * 32];
    float s = 0.0f, ss = 0.0f;
#pragma unroll
    for (int j = 0; j < 32; ++j) { float v = g[j]; s += v; ss += v * v; }
    const float mu  = s * (1.0f / 32.0f);
    const float var = ss * (1.0f / 32.0f) - mu * mu;
    const float rs  = rsqrtf(var + 1e-5f);
    const int node  = node0 + row;
#pragma unroll
    for (int j = 0; j < 32; ++j) {
      int ch = grp * 32 + j;
      hout[node * 64 + ch] = (g[j] - mu) * rs * gnw[ch] + gnb[ch];
    }
  }
}

// ------------------------------ decoder ------------------------------------
__global__ void __launch_bounds__(32)
k_decoder(const float* __restrict__ h,
          const v2f* __restrict__ pW1, const float* __restrict__ b1,
          const float* __restrict__ W2, const float* __restrict__ b2,
          float* __restrict__ out) {
  __shared__ float At[16 * 68];
  __shared__ float Ht[16 * 68];
  const int lane  = threadIdx.x;
  const int node0 = blockIdx.x * 16;
  for (int i = lane; i < 16 * 68; i += 32) {
    int r = i / 68, c = i - r * 68;
    At[i] = (c < 64) ? h[(node0 + r) * 64 + c] : 0.0f;
  }
  __syncthreads();
  v8f acc[1][4];
  mm16x64_mt<64, 68, 1>(At, 0, pW1, acc);
  epi_to_lds(acc[0], b1, Ht, 68, true);
  __syncthreads();
  for (int p = lane; p < 64; p += 32) {
    int row = p >> 2, col = p & 3;
    float s = b2[col];
#pragma unroll 8
    for (int k = 0; k < 64; ++k) s += Ht[row * 68 + k] * W2[k * 4 + col];
    out[(node0 + row) * 4 + col] = s;
  }
}

// ---------------------------------------------------------------------------
static inline int k4of(int fan_in) { return (fan_in + 3) / 4; }

extern "C" void kernel_launch(void* const* d_in, const int* in_sizes, int n_in,
                              void* d_out, int out_size, void* d_ws,
                              size_t ws_size, hipStream_t stream) {
  (void)out_size; (void)ws_size;
  const float *x, *ea;
  const int* ei;
  const float *encW1, *encb1, *encW2, *encb2;
  const float *phiW1[2], *phib1[2], *phiW2[2], *phib2[2];
  const float *gamW1[2], *gamb1[2], *gamW2[2], *gamb2[2];
  const float *decW1, *decb1, *decW2, *decb2, *gnw, *gnb;
  auto F = [&](int i) { return (const float*)d_in[i]; };

  if (n_in >= 29 && in_sizes[0] == N_NODES * 9) {
    // dict insertion order
    x = F(0); ei = (const int*)d_in[1]; ea = F(2);
    encW1 = F(3); encb1 = F(4); encW2 = F(5); encb2 = F(6);
    for (int l = 0; l < 2; ++l) {
      int b = 7 + l * 8;
      phiW1[l] = F(b);     phib1[l] = F(b + 1);
      phiW2[l] = F(b + 2); phib2[l] = F(b + 3);
      gamW1[l] = F(b + 4); gamb1[l] = F(b + 5);
      gamW2[l] = F(b + 6); gamb2[l] = F(b + 7);
    }
    decW1 = F(23); decb1 = F(24); decW2 = F(25); decb2 = F(26);
    gnw = F(27); gnb = F(28);
  } else {
    // jax pytree (sorted-key) order
    ea = F(0); ei = (const int*)d_in[1];
    decW1 = F(2); decW2 = F(3); decb1 = F(4); decb2 = F(5);
    encW1 = F(6); encW2 = F(7); encb1 = F(8); encb2 = F(9);
    gnb = F(10); gnw = F(11);
    for (int l = 0; l < 2; ++l) {
      int b = 12 + l * 8;
      gamW1[l] = F(b);     gamW2[l] = F(b + 1);
      gamb1[l] = F(b + 2); gamb2[l] = F(b + 3);
      phiW1[l] = F(b + 4); phiW2[l] = F(b + 5);
      phib1[l] = F(b + 6); phib2[l] = F(b + 7);
    }
    x = F(28);
  }

  // workspace: h_ping | h_pong | aggr | cnt | packed weights  (~39 MB)
  float* h0   = (float*)d_ws;
  float* h1   = h0 + (size_t)N_NODES * 64;
  float* aggr = h1 + (size_t)N_NODES * 64;
  float* cnt  = aggr + (size_t)N_NODES * 64;
  float* pk   = cnt + N_NODES;   // even offset -> 8B aligned for v2f

  auto pack = [&](const float* W, int fan_in) {
    int k4 = k4of(fan_in);
    int tot = k4 * 128;
    float* dst = pk;
    k_pack<<<(tot + 255) / 256, 256, 0, stream>>>(W, dst, fan_in, k4);
    pk += (size_t)tot * 2;
    return (const v2f*)dst;
  };

  const v2f* pEnc1 = pack(encW1, 9);
  const v2f* pEnc2 = pack(encW2, 64);
  const v2f *pPhi1[2], *pPhi2[2], *pGam1[2], *pGam2[2];
  for (int l = 0; l < 2; ++l) {
    pPhi1[l] = pack(phiW1[l], 131);
    pPhi2[l] = pack(phiW2[l], 64);
    pGam1[l] = pack(gamW1[l], 128);
    pGam2[l] = pack(gamW2[l], 64);
  }
  const v2f* pDec1 = pack(decW1, 64);

  (void)hipMemsetAsync(cnt, 0, N_NODES * sizeof(float), stream);
  k_count<<<(N_EDGES + 255) / 256, 256, 0, stream>>>(ei + N_EDGES, cnt);
  k_inv<<<(N_NODES + 255) / 256, 256, 0, stream>>>(cnt);

  k_encoder<<<N_NODES / 16, 32, 0, stream>>>(x, pEnc1, encb1, pEnc2, encb2, h0);

  float* hc = h0;
  float* hn = h1;
  for (int l = 0; l < 2; ++l) {
    (void)hipMemsetAsync(aggr, 0, (size_t)N_NODES * 64 * sizeof(float), stream);
    k_edge<<<N_EDGES / (EDGE_R * 16), 32, 0, stream>>>(
        hc, ei, ea, pPhi1[l], phib1[l], pPhi2[l], phib2[l], aggr);
    k_update<<<N_NODES / 16, 32, 0, stream>>>(hc, aggr, cnt, pGam1[l], gamb1[l],
                                              pGam2[l], gamb2[l], gnw, gnb, hn);
    float* t = hc; hc = hn; hn = t;
  }

  k_decoder<<<N_NODES / 16, 32, 0, stream>>>(hc, pDec1, decb1, decW2, decb2,
                                             (float*)d_out);
}